// GCN_46179488367199
// MI455X (gfx1250) — compile-verified
//
#include <hip/hip_runtime.h>
#include <hip/hip_bf16.h>

typedef __attribute__((ext_vector_type(2))) float v2f;
typedef __attribute__((ext_vector_type(8))) float v8f;

#define HID 64
#define LDSW 66   // padded LDS stride (floats): spreads ds banks, keeps 8B-aligned pairs

// ---------------------------------------------------------------------------
// C[M,64] = op(A)[M,64] @ W[64,64] (+ bias), op = ReLU iff RELU_A (compile-time,
// so the no-activation path has zero extra VALU and the ReLU path has no selects).
// One wave per 16-row strip; four 16x16 N-tiles in v8f accumulators; K stepped
// by 4 via V_WMMA_F32_16X16X4_F32 (wave32 fp32 matrix path).
// W staged TRANSPOSED into LDS once per block: Wt[n*LDSW + k] = W[k][n] -> each
// B fragment {B[k][n], B[k+1][n]} is one aligned ds_load_b64 (compiler pairs
// them into ds_load_2addr_b64 with immediate offsets off one base per N-tile).
// A fragment (ISA 7.12.2): lane L -> M=L%16, v.x = K=k0+2*(L/16), v.y = K+1.
// C/D: VGPR r -> M = r + 8*(L/16), N = L%16.
// ---------------------------------------------------------------------------
template <int RELU_A>
__global__ void gcn_gemm64_wmma(const float* __restrict__ A,
                                const float* __restrict__ W,
                                const float* __restrict__ bias,
                                float* __restrict__ C, int M) {
  __shared__ float Wt[HID * LDSW];

  // Stage W transposed: coalesced global reads, stride-2-bank LDS writes.
  for (int idx = threadIdx.x; idx < HID * HID; idx += blockDim.x) {
    const int k = idx >> 6;
    const int n = idx & 63;
    Wt[n * LDSW + k] = W[idx];
  }
  __syncthreads();

  const int lane = threadIdx.x & 31;
  const int wave = threadIdx.x >> 5;
  const int strip = blockIdx.x * (blockDim.x >> 5) + wave;
  const int m0 = strip * 16;
  if (m0 >= M) return;                 // wave-uniform: EXEC stays all-1 for WMMA

  const int mrow = lane & 15;          // M (for A) / N (for B,C)
  const int kh   = lane >> 4;          // K-half select

  const float* Arow = A + (size_t)(m0 + mrow) * HID;

  v8f acc[4] = {};

#pragma unroll
  for (int k0 = 0; k0 < HID; k0 += 4) {
    v2f a = *(const v2f*)(Arow + k0 + 2 * kh);
    if constexpr (RELU_A) {
      a.x = fmaxf(a.x, 0.0f);
      a.y = fmaxf(a.y, 0.0f);
    }
#pragma unroll
    for (int nt = 0; nt < 4; ++nt) {
      const int col = nt * 16 + mrow;
      v2f b = *(const v2f*)(&Wt[col * LDSW + k0 + 2 * kh]);   // ds_load_b64
      acc[nt] = __builtin_amdgcn_wmma_f32_16x16x4_f32(
          /*neg_a=*/false, a, /*neg_b=*/false, b,
          /*c_mod=*/(short)0, acc[nt], /*reuse_a=*/false, /*reuse_b=*/false);
    }
  }

#pragma unroll
  for (int nt = 0; nt < 4; ++nt) {
    const int nn = nt * 16 + mrow;
    const float bv = bias ? bias[nn] : 0.0f;
#pragma unroll
    for (int r = 0; r < 8; ++r) {
      const int mm = m0 + r + 8 * kh;
      C[(size_t)mm * HID + nn] = acc[nt][r] + bv;
    }
  }
}

// ---------------------------------------------------------------------------
// Degree / normalization (GCN symmetric norm with self loops).
// ---------------------------------------------------------------------------
__global__ void gcn_deg_init(float* __restrict__ deg, int M) {
  int i = blockIdx.x * blockDim.x + threadIdx.x;
  if (i < M) deg[i] = 1.0f;            // self loop
}

__global__ void gcn_deg_count(const int* __restrict__ dst, float* __restrict__ deg, int E) {
  int e = blockIdx.x * blockDim.x + threadIdx.x;
  if (e < E) unsafeAtomicAdd(&deg[dst[e]], 1.0f);   // hw global_atomic_add_f32
}

__global__ void gcn_deg_finish(float* __restrict__ dinv, int M) {
  int i = blockIdx.x * blockDim.x + threadIdx.x;
  if (i < M) {
    float d = dinv[i];
    dinv[i] = (d > 0.0f) ? (1.0f / sqrtf(d)) : 0.0f;
  }
}

// ---------------------------------------------------------------------------
// out[i] = hw[i]*dinv[i]^2 (self-loop) + bg + op(h_res[i]); op = ReLU iff
// RELU_RES (compile-time).  Stored h is pre-activation; ReLU fused into reads.
// ---------------------------------------------------------------------------
template <int RELU_RES>
__global__ void gcn_self_bias_res(const float* __restrict__ hw,
                                  const float* __restrict__ hres,
                                  const float* __restrict__ dinv,
                                  const float* __restrict__ bg,
                                  float* __restrict__ out, int M) {
  int t = blockIdx.x * blockDim.x + threadIdx.x;
  if (t >= M * 16) return;
  int i = t >> 4;
  int c = (t & 15) * 4;
  float di = dinv[i];
  float w = di * di;
  const float4 hw4 = *(const float4*)(hw + (size_t)i * HID + c);
  float4 hr4 = *(const float4*)(hres + (size_t)i * HID + c);
  if constexpr (RELU_RES) {
    hr4.x = fmaxf(hr4.x, 0.0f);
    hr4.y = fmaxf(hr4.y, 0.0f);
    hr4.z = fmaxf(hr4.z, 0.0f);
    hr4.w = fmaxf(hr4.w, 0.0f);
  }
  const float4 bg4 = *(const float4*)(bg + c);
  float4 o;
  o.x = fmaf(hw4.x, w, bg4.x + hr4.x);
  o.y = fmaf(hw4.y, w, bg4.y + hr4.y);
  o.z = fmaf(hw4.z, w, bg4.z + hr4.z);
  o.w = fmaf(hw4.w, w, bg4.w + hr4.w);
  *(float4*)(out + (size_t)i * HID + c) = o;
}

// ---------------------------------------------------------------------------
// Edge scatter: out[dst] += hw[src] * dinv[src]*dinv[dst].
// One wave per edge, 2 channels per lane: float2 gather (global_load_b64) +
// two hardware fp32 atomics resolving in L2 (whole h fits in 192MB L2).
// ---------------------------------------------------------------------------
__global__ void gcn_edge_scatter(const int* __restrict__ src,
                                 const int* __restrict__ dst,
                                 const float* __restrict__ dinv,
                                 const float* __restrict__ hw,
                                 float* __restrict__ out, int E) {
  long t = (long)blockIdx.x * blockDim.x + threadIdx.x;
  if (t >= (long)E * 32) return;
  int e = (int)(t >> 5);
  int c = (int)(t & 31) * 2;
  int s = src[e];
  int d = dst[e];
  float nrm = dinv[s] * dinv[d];
  const float2 v = *(const float2*)(hw + (size_t)s * HID + c);
  unsafeAtomicAdd(out + (size_t)d * HID + c,     v.x * nrm);
  unsafeAtomicAdd(out + (size_t)d * HID + c + 1, v.y * nrm);
}

// ---------------------------------------------------------------------------
// Fused MLP readout 64 -> 32 -> 16 -> 1, one thread per node.
// (Final h has no ReLU per the reference: last GCN layer skips activation.)
// ---------------------------------------------------------------------------
__global__ void gcn_readout(const float* __restrict__ h,
                            const float* __restrict__ W0, const float* __restrict__ b0,
                            const float* __restrict__ W1, const float* __restrict__ b1,
                            const float* __restrict__ W2, const float* __restrict__ b2,
                            float* __restrict__ y, int M) {
  int i = blockIdx.x * blockDim.x + threadIdx.x;
  if (i >= M) return;
  float hr[HID];
  const float* hi = h + (size_t)i * HID;
#pragma unroll
  for (int k = 0; k < HID; ++k) hr[k] = hi[k];

  float t0[32];
#pragma unroll 4
  for (int j = 0; j < 32; ++j) {
    float s = b0[j];
    for (int k = 0; k < HID; ++k) s = fmaf(hr[k], W0[k * 32 + j], s);
    t0[j] = fmaxf(s, 0.0f);
  }
  float t1[16];
#pragma unroll 4
  for (int j = 0; j < 16; ++j) {
    float s = b1[j];
    for (int k = 0; k < 32; ++k) s = fmaf(t0[k], W1[k * 16 + j], s);
    t1[j] = fmaxf(s, 0.0f);
  }
  float s = b2[0];
#pragma unroll
  for (int k = 0; k < 16; ++k) s = fmaf(t1[k], W2[k], s);
  y[i] = s;
}

// ---------------------------------------------------------------------------
extern "C" void kernel_launch(void* const* d_in, const int* in_sizes, int n_in,
                              void* d_out, int out_size, void* d_ws, size_t ws_size,
                              hipStream_t stream) {
  const float* x     = (const float*)d_in[0];
  const int*   ei    = (const int*)d_in[1];
  const float* W_enc = (const float*)d_in[2];
  const float* b_enc = (const float*)d_in[3];
  const float* Wg    = (const float*)d_in[4];
  const float* bg    = (const float*)d_in[5];
  const float* W0    = (const float*)d_in[6];
  const float* b0    = (const float*)d_in[7];
  const float* W1    = (const float*)d_in[8];
  const float* b1    = (const float*)d_in[9];
  const float* W2    = (const float*)d_in[10];
  const float* b2    = (const float*)d_in[11];
  float* y = (float*)d_out;

  const int M = in_sizes[0] / HID;       // 100000 nodes
  const int E = in_sizes[1] / 2;         // 1600000 edges
  const int* src = ei;
  const int* dst = ei + E;

  // Workspace layout (~77.2 MB fp32; whole graph stays L2-resident):
  float* hA   = (float*)d_ws;                    // M*64
  float* hw   = hA + (size_t)M * HID;            // M*64
  float* hB   = hw + (size_t)M * HID;            // M*64
  float* dinv = hB + (size_t)M * HID;            // M

  const int B = 256;

  // --- degree + symmetric normalization -----------------------------------
  gcn_deg_init  <<<(M + B - 1) / B, B, 0, stream>>>(dinv, M);
  gcn_deg_count <<<(E + B - 1) / B, B, 0, stream>>>(dst, dinv, E);
  gcn_deg_finish<<<(M + B - 1) / B, B, 0, stream>>>(dinv, M);

  // --- encoder GEMM: hA = x @ W_enc + b_enc (no input activation) ---------
  const int strips = (M + 15) / 16;
  const int gemmBlocks = (strips + 7) / 8;       // 8 waves (16-row strips) / block
  gcn_gemm64_wmma<0><<<gemmBlocks, 256, 0, stream>>>(x, W_enc, b_enc, hA, M);

  // --- 4 GCN layers (h buffers hold PRE-activation; ReLU fused into reads) -
  float* h = hA;
  float* o = hB;
  for (int l = 0; l < 4; ++l) {
    const float* Wl  = Wg + (size_t)l * HID * HID;
    const float* bgl = bg + (size_t)l * HID;
    const int nEp = (int)(((long)M * 16 + B - 1) / B);
    if (l >= 1) {                                 // layers 0..2 ended with ReLU
      gcn_gemm64_wmma<1><<<gemmBlocks, 256, 0, stream>>>(h, Wl, nullptr, hw, M);
      gcn_self_bias_res<1><<<nEp, B, 0, stream>>>(hw, h, dinv, bgl, o, M);
    } else {
      gcn_gemm64_wmma<0><<<gemmBlocks, 256, 0, stream>>>(h, Wl, nullptr, hw, M);
      gcn_self_bias_res<0><<<nEp, B, 0, stream>>>(hw, h, dinv, bgl, o, M);
    }
    gcn_edge_scatter<<<(int)(((long)E * 32 + B - 1) / B), B, 0, stream>>>(
        src, dst, dinv, hw, o, E);
    float* tmp = h; h = o; o = tmp;              // ping-pong
  }

  // --- fused MLP readout (layer 3 had no ReLU; read h raw) ----------------
  gcn_readout<<<(M + B - 1) / B, B, 0, stream>>>(h, W0, b0, W1, b1, W2, b2, y, M);
}